// DynamicNodeMask_36679020708615
// MI455X (gfx1250) — compile-verified
//
#include <hip/hip_runtime.h>
#include <stdint.h>

typedef float        v4f  __attribute__((ext_vector_type(4)));
typedef unsigned int u32x4 __attribute__((ext_vector_type(4)));
typedef int          i32x4 __attribute__((ext_vector_type(4)));
typedef int          i32x8 __attribute__((ext_vector_type(8)));

#define E_DIM 1024
#define BLOCK 256

#if __has_builtin(__builtin_amdgcn_tensor_load_to_lds)
#define HAVE_TDM 1
#else
#define HAVE_TDM 0
#endif

__device__ __forceinline__ uint32_t pcg_hash(uint32_t v) {
  uint32_t s = v * 747796405u + 2891336453u;
  uint32_t w = ((s >> ((s >> 28u) + 4u)) ^ s) * 277803737u;
  return (w >> 22u) ^ w;
}

__global__ __launch_bounds__(BLOCK) void DynamicNodeMask_kernel(
    const float* __restrict__ embeds, const float* __restrict__ factors,
    const float* __restrict__ mask_token, float* __restrict__ out_embeds,
    float* __restrict__ out_mask) {
  __shared__ __align__(16) float s_row[E_DIM];   // embeds row staged by TDM
  __shared__ uint32_t s_scores[E_DIM];
  __shared__ uint32_t s_hist[E_DIM];             // 1024 bins on score[31:22]
  __shared__ uint32_t s_wsum[BLOCK / 32];
  __shared__ uint32_t s_selbin, s_selr;

  const uint32_t row = blockIdx.x;
  const uint32_t tid = threadIdx.x;
  const uint32_t c0  = tid * 4u;
  const float* grow  = embeds + (size_t)row * E_DIM;

  // ---- Stage embeds row into LDS via Tensor Data Mover (async, TENSORcnt) ----
#if HAVE_TDM
  if (tid == 0) {
    uint64_t ga  = (uint64_t)(uintptr_t)grow;
    uint32_t lds = (uint32_t)(uintptr_t)(void*)s_row;  // low 32b of flat addr = LDS byte offset
    // D# group0: count=1 | lds_addr | global_addr[56:0] | type=2
    u32x4 g0 = {1u, lds, (uint32_t)ga,
                (uint32_t)((ga >> 32) & 0x01FFFFFFu) | 0x80000000u};
    // D# group1: data_size=4B; tensor_dim0=1024; tensor_dim1=1; tile_dim0=1024;
    //            tile_dim1=1; tensor_dim0_stride=1024; tensor_dim1_stride=1024
    i32x8 g1 = {(int)0x00020000, (int)(E_DIM << 16), (int)0x00010000,
                (int)(E_DIM << 16), 1, E_DIM, (int)(E_DIM << 16), 0};
    i32x4 gz = {0, 0, 0, 0};
#if __has_include(<hip/amd_detail/amd_gfx1250_TDM.h>)
    i32x8 gz8 = {0, 0, 0, 0, 0, 0, 0, 0};
    __builtin_amdgcn_tensor_load_to_lds(g0, g1, gz, gz, gz8, 0);   // clang-23 6-arg
#else
    __builtin_amdgcn_tensor_load_to_lds(g0, g1, gz, gz, 0);        // ROCm 7.2 5-arg
#endif
  }
#else
  {  // fallback: cooperative synchronous staging
    v4f e = *(const v4f*)(grow + c0);
    *(v4f*)&s_row[c0] = e;
  }
#endif

  // ---- per-row masked count k = max(floor(307.2 * factor), 1) ----
  const float f = factors[row];
  int k_i = (int)(307.2f * f);
  uint32_t k = (uint32_t)(k_i < 1 ? 1 : k_i);
  if (k > E_DIM) k = E_DIM;

  // ---- generate scores, clear histogram ----
  uint32_t sc[4];
#pragma unroll
  for (int j = 0; j < 4; ++j) {
    sc[j] = pcg_hash(row * (uint32_t)E_DIM + c0 + (uint32_t)j);
    s_scores[c0 + j] = sc[j];
  }
#pragma unroll
  for (int j = 0; j < 4; ++j) s_hist[tid + j * BLOCK] = 0u;
  __syncthreads();

#pragma unroll
  for (int j = 0; j < 4; ++j) atomicAdd(&s_hist[sc[j] >> 22u], 1u);
  __syncthreads();

  // ---- hierarchical exclusive prefix over 1024 bins (wave32 shfl scan) ----
  uint32_t h[4], tsum = 0u;
#pragma unroll
  for (int j = 0; j < 4; ++j) { h[j] = s_hist[c0 + j]; tsum += h[j]; }
  const uint32_t lane = tid & 31u, wv = tid >> 5u;
  uint32_t incl = tsum;
#pragma unroll
  for (int d = 1; d < 32; d <<= 1) {
    uint32_t n = __shfl_up(incl, (unsigned)d, 32);
    if (lane >= (uint32_t)d) incl += n;
  }
  if (lane == 31u) s_wsum[wv] = incl;
  __syncthreads();
  uint32_t wpre = 0u;
#pragma unroll
  for (uint32_t w = 0; w < BLOCK / 32; ++w) wpre += (w < wv) ? s_wsum[w] : 0u;
  uint32_t run = wpre + incl - tsum;  // exclusive prefix before this thread's 4 bins
#pragma unroll
  for (int j = 0; j < 4; ++j) {
    uint32_t c = h[j];
    if (run < k && run + c >= k) { s_selbin = c0 + (uint32_t)j; s_selr = k - run; }
    run += c;
  }
  __syncthreads();
  const uint32_t selbin = s_selbin, selr = s_selr;

  // ---- wait for TDM DMA of the row, then apply mask ----
#if HAVE_TDM
  if (tid == 0) {
#if __has_builtin(__builtin_amdgcn_s_wait_tensorcnt)
    __builtin_amdgcn_s_wait_tensorcnt(0);
#else
    asm volatile("s_wait_tensorcnt 0x0" ::: "memory");
#endif
  }
  __syncthreads();
  asm volatile("" ::: "memory");  // TDM wrote LDS behind the compiler's back
#endif

  const v4f tok = *(const v4f*)(mask_token + c0);  // 4KB row, L2-resident
  const v4f e   = *(const v4f*)&s_row[c0];
  v4f oe, om;
#pragma unroll
  for (int j = 0; j < 4; ++j) {
    const uint32_t v = sc[j];
    const uint32_t top = v >> 22u;
    bool masked;
    if (top < selbin) {
      masked = true;
    } else if (top > selbin) {
      masked = false;
    } else {  // rare path: resolve within-bin rank (expected ~1 element/row)
      uint32_t r = 0;
      const uint32_t myidx = c0 + (uint32_t)j;
      for (uint32_t m = 0; m < E_DIM; ++m) {
        uint32_t u = s_scores[m];
        if ((u >> 22u) == selbin && (u < v || (u == v && m < myidx))) r++;
      }
      masked = (r < selr);
    }
    om[j] = masked ? 0.0f : 1.0f;
    oe[j] = masked ? tok[j] : e[j];
  }

  const size_t base = (size_t)row * E_DIM + c0;
  __builtin_nontemporal_store(oe, (v4f*)(out_embeds + base));  // write-once stream
  __builtin_nontemporal_store(om, (v4f*)(out_mask + base));
}

extern "C" void kernel_launch(void* const* d_in, const int* in_sizes, int n_in,
                              void* d_out, int out_size, void* d_ws, size_t ws_size,
                              hipStream_t stream) {
  (void)n_in; (void)out_size; (void)d_ws; (void)ws_size;
  const float* embeds  = (const float*)d_in[0];
  const float* factors = (const float*)d_in[1];
  const float* token   = (const float*)d_in[2];
  const int B = in_sizes[1];                 // dynamic_factors has B elements
  float* out_embeds = (float*)d_out;
  float* out_mask   = (float*)d_out + (size_t)B * E_DIM;
  DynamicNodeMask_kernel<<<dim3((unsigned)B), dim3(BLOCK), 0, stream>>>(
      embeds, factors, token, out_embeds, out_mask);
}